// Aggregator_12429635354865
// MI455X (gfx1250) — compile-verified
//
#include <hip/hip_runtime.h>

#define D 128
#define LN_EPS 1e-5f
#define NEG_SLOPE 0.01f

typedef float v2f __attribute__((ext_vector_type(2)));
typedef float v8f __attribute__((ext_vector_type(8)));

// ---------------------------------------------------------------------------
// Zero the side/accumulator buffer (needed every call: atomics accumulate).
// ---------------------------------------------------------------------------
__global__ void zero_f4(float4* __restrict__ p, int n4) {
  int i = blockIdx.x * blockDim.x + threadIdx.x;
  if (i < n4) p[i] = make_float4(0.f, 0.f, 0.f, 0.f);
}

// ---------------------------------------------------------------------------
// Repack W1/W2 (row-major 128x128) into exact WMMA B-fragment order so the
// GEMM inner loop issues one contiguous global_load_b64 per matrix per k-step.
// Fragment layout for V_WMMA_F32_16X16X4_F32 B (4x16):
//   lanes 0-15 hold K = 4*ks+{0,1}, lanes 16-31 hold K = 4*ks+{2,3}, N = lane%16.
// Wt[m*8192 + ks*256 + nb*32 + lane] = { W[k][n], W[k+1][n] }.
// ---------------------------------------------------------------------------
__global__ void pack_w(const float* __restrict__ W1, const float* __restrict__ W2,
                       float2* __restrict__ Wt) {
  int t = blockIdx.x * blockDim.x + threadIdx.x;   // 0 .. 16383
  int m    = t >> 13;                              // matrix select
  int rest = t & 8191;
  int ks   = rest >> 8;                            // 0..31
  int nb   = (rest >> 5) & 7;                      // 0..7
  int lane = rest & 31;
  int k = ks * 4 + ((lane >> 4) << 1);
  int n = nb * 16 + (lane & 15);
  const float* W = m ? W2 : W1;
  Wt[t] = make_float2(W[k * D + n], W[(k + 1) * D + n]);
}

// ---------------------------------------------------------------------------
// COO SpMM: one wave32 per edge; lane handles 4 columns (float4 gather from
// ego[col], scale by val, atomicAdd into side[row]). ego (51MB) is L2-resident
// on MI455X (192MB L2), so the random gathers and the float atomics stay in L2.
// ---------------------------------------------------------------------------
__global__ void spmm_coo(const float* __restrict__ ego,
                         const int* __restrict__ erow,
                         const int* __restrict__ ecol,
                         const float* __restrict__ eval,
                         float* __restrict__ side, int E) {
  int gid  = blockIdx.x * blockDim.x + threadIdx.x;
  int edge = gid >> 5;
  int lane = gid & 31;
  if (edge >= E) return;
  int   r = erow[edge];
  int   c = ecol[edge];
  float v = eval[edge];
  const float4* src = (const float4*)(ego + (size_t)c * D);
  float4 m = src[lane];
  float* dst = side + (size_t)r * D + lane * 4;
  atomicAdd(dst + 0, v * m.x);
  atomicAdd(dst + 1, v * m.y);
  atomicAdd(dst + 2, v * m.z);
  atomicAdd(dst + 3, v * m.w);
}

// ---------------------------------------------------------------------------
// Fused: X1 = ego+side, X2 = ego*side; Y1 = leaky(X1@W1+b1), Y2 = leaky(X2@W2+b2)
// out = LN(Y1)*g1+be1 + LN(Y2)*g2+be2.
// One block = 16 rows. 8 waves; wave nb computes the 16x16 output tile at
// columns [nb*16, nb*16+16) for BOTH branches via V_WMMA_F32_16X16X4_F32
// (full f32 precision; K-loop of 32 steps, 2 WMMAs per step).
// ---------------------------------------------------------------------------
__global__ __launch_bounds__(256) void fused_gemm_ln(
    const float* __restrict__ ego, const float* __restrict__ side,
    const float2* __restrict__ Wt,
    const float* __restrict__ b1,  const float* __restrict__ b2,
    const float* __restrict__ g1,  const float* __restrict__ be1,
    const float* __restrict__ g2,  const float* __restrict__ be2,
    float* __restrict__ out) {
  __shared__ float X1[16][D];
  __shared__ float X2[16][D];
  __shared__ float Y1[16][D];
  __shared__ float Y2[16][D];

  const int tid  = threadIdx.x;
  const int row0 = blockIdx.x * 16;

  // Phase 1: build X1 = ego+side, X2 = ego*side (16 x 128) in LDS.
  const float4* e4 = (const float4*)ego;
  const float4* s4 = (const float4*)side;
  for (int i = tid; i < 16 * (D / 4); i += 256) {
    int r  = i >> 5;          // D/4 == 32 float4 per row
    int c4 = i & 31;
    float4 e = e4[(size_t)(row0 + r) * (D / 4) + c4];
    float4 s = s4[(size_t)(row0 + r) * (D / 4) + c4];
    int c = c4 * 4;
    X1[r][c + 0] = e.x + s.x;  X1[r][c + 1] = e.y + s.y;
    X1[r][c + 2] = e.z + s.z;  X1[r][c + 3] = e.w + s.w;
    X2[r][c + 0] = e.x * s.x;  X2[r][c + 1] = e.y * s.y;
    X2[r][c + 2] = e.z * s.z;  X2[r][c + 3] = e.w * s.w;
  }
  __syncthreads();

  // Phase 2: WMMA. A-frag (16x4 f32): lanes 0-15 hold K=0,1; lanes 16-31 K=2,3.
  // B-frag comes pre-packed from Wt (one b64 load per matrix per k-step).
  const int lane = tid & 31;
  const int nb   = tid >> 5;                 // wave id -> N tile
  const int mrow = lane & 15;                // A row / B col within tile
  const int kk   = (lane >> 4) << 1;         // 0 or 2
  const int nIdx = nb * 16 + mrow;           // global output column
  const float2* wt1 = Wt + nb * 32 + lane;   // matrix 1 fragment stream
  const float2* wt2 = wt1 + 8192;            // matrix 2 fragment stream
  v8f acc1 = {0.f, 0.f, 0.f, 0.f, 0.f, 0.f, 0.f, 0.f};
  v8f acc2 = acc1;
  for (int ks = 0; ks < D / 4; ++ks) {
    const int kb = ks * 4 + kk;
    v2f a1;  a1.x = X1[mrow][kb];  a1.y = X1[mrow][kb + 1];
    v2f a2;  a2.x = X2[mrow][kb];  a2.y = X2[mrow][kb + 1];
    float2 w1f = wt1[ks * 256];
    float2 w2f = wt2[ks * 256];
    v2f bw1; bw1.x = w1f.x; bw1.y = w1f.y;
    v2f bw2; bw2.x = w2f.x; bw2.y = w2f.y;
    acc1 = __builtin_amdgcn_wmma_f32_16x16x4_f32(false, a1, false, bw1,
                                                 (short)0, acc1, false, false);
    acc2 = __builtin_amdgcn_wmma_f32_16x16x4_f32(false, a2, false, bw2,
                                                 (short)0, acc2, false, false);
  }

  // Phase 3: bias + LeakyReLU -> LDS staging for LayerNorm.
  const float bb1 = b1[nIdx], bb2 = b2[nIdx];
  const int mbase = (lane >> 4) << 3;        // 0 or 8
  #pragma unroll
  for (int r = 0; r < 8; ++r) {
    float y1 = acc1[r] + bb1;
    y1 = y1 > 0.f ? y1 : y1 * NEG_SLOPE;
    Y1[mbase + r][nIdx] = y1;
    float y2 = acc2[r] + bb2;
    y2 = y2 > 0.f ? y2 : y2 * NEG_SLOPE;
    Y2[mbase + r][nIdx] = y2;
  }
  __syncthreads();

  // Phase 4: per-row LayerNorm for both branches, combine, write out.
  // 16 threads per row (tid>>4 = row), 8 columns each; xor-shuffle reduction
  // over the 16-lane half-wave (masks 8..1 never cross the 16-lane boundary).
  const int r  = tid >> 4;
  const int cg = tid & 15;
  float s1 = 0.f, q1 = 0.f, s2 = 0.f, q2 = 0.f;
  #pragma unroll
  for (int j = 0; j < 8; ++j) {
    float v = Y1[r][cg * 8 + j]; s1 += v; q1 += v * v;
    v = Y2[r][cg * 8 + j];       s2 += v; q2 += v * v;
  }
  #pragma unroll
  for (int off = 8; off > 0; off >>= 1) {
    s1 += __shfl_xor(s1, off, 32);
    q1 += __shfl_xor(q1, off, 32);
    s2 += __shfl_xor(s2, off, 32);
    q2 += __shfl_xor(q2, off, 32);
  }
  const float mu1 = s1 * (1.f / D);
  const float mu2 = s2 * (1.f / D);
  const float rs1 = rsqrtf(q1 * (1.f / D) - mu1 * mu1 + LN_EPS);
  const float rs2 = rsqrtf(q2 * (1.f / D) - mu2 * mu2 + LN_EPS);
  float* orow = out + (size_t)(row0 + r) * D;
  #pragma unroll
  for (int j = 0; j < 8; ++j) {
    int c = cg * 8 + j;
    orow[c] = (Y1[r][c] - mu1) * rs1 * g1[c] + be1[c]
            + (Y2[r][c] - mu2) * rs2 * g2[c] + be2[c];
  }
}

// ---------------------------------------------------------------------------
extern "C" void kernel_launch(void* const* d_in, const int* in_sizes, int n_in,
                              void* d_out, int out_size, void* d_ws, size_t ws_size,
                              hipStream_t stream) {
  const float* ego  = (const float*)d_in[0];
  const int*   erow = (const int*)  d_in[1];
  const int*   ecol = (const int*)  d_in[2];
  const float* eval = (const float*)d_in[3];
  const float* W1   = (const float*)d_in[4];
  const float* b1   = (const float*)d_in[5];
  const float* W2   = (const float*)d_in[6];
  const float* b2   = (const float*)d_in[7];
  const float* g1   = (const float*)d_in[8];
  const float* be1  = (const float*)d_in[9];
  const float* g2   = (const float*)d_in[10];
  const float* be2  = (const float*)d_in[11];
  float* out  = (float*)d_out;

  const int N = in_sizes[0] / D;        // 100000
  const int E = in_sizes[1];            // 1600000

  // Workspace layout: side [N*128 f32] then packed weights [2*8192 float2].
  float*  side = (float*)d_ws;
  float2* Wt   = (float2*)(side + (size_t)N * D);

  // 1) side = 0
  const int n4 = N * (D / 4);
  zero_f4<<<(n4 + 255) / 256, 256, 0, stream>>>((float4*)side, n4);

  // 2) repack W1/W2 into WMMA B-fragment order (16384 float2 -> 64 blocks)
  pack_w<<<64, 256, 0, stream>>>(W1, W2, Wt);

  // 3) side += A_in @ ego  (one wave32 per edge, 8 edges per 256-thread block)
  const long long threads = (long long)E * 32;
  spmm_coo<<<(int)((threads + 255) / 256), 256, 0, stream>>>(
      ego, erow, ecol, eval, side, E);

  // 4) fused bi-interaction GEMMs + bias + LeakyReLU + LayerNorm + add
  fused_gemm_ln<<<N / 16, 256, 0, stream>>>(
      ego, side, Wt, b1, b2, g1, be1, g2, be2, out);
}